// Networkkkk_54065048322604
// MI455X (gfx1250) — compile-verified
//
#include <hip/hip_runtime.h>
#include <math.h>

#define Bz   64
#define Rn   200
#define Kh   8
#define IND  200
#define D1   32
#define D2   32
#define D3   512
#define D4   256
#define D5   8
#define Tt   2

typedef float v2f __attribute__((ext_vector_type(2)));
typedef float v8f __attribute__((ext_vector_type(8)));

__device__ __forceinline__ v8f wmma4(v2f a, v2f b, v8f c) {
  // D = A(16x4,f32) * B(4x16,f32) + C(16x16,f32), exact fp32 accumulate
  return __builtin_amdgcn_wmma_f32_16x16x4_f32(false, a, false, b, (short)0, c, false, false);
}

// ---- hypernetwork: W[r,m] = sum_k relu(Wa[r,k]) * Wb[k,m] + b[m] ----
__global__ void hyper_kernel(const float* __restrict__ Wa, const float* __restrict__ Wb,
                             const float* __restrict__ bb, float* __restrict__ out, int M) {
  int idx = blockIdx.x * blockDim.x + threadIdx.x;
  if (idx >= Rn * M) return;
  int r = idx / M, m = idx % M;
  float acc = bb[m];
#pragma unroll
  for (int k = 0; k < Kh; ++k) {
    float a = Wa[r * Kh + k];
    a = a > 0.f ? a : 0.f;
    acc = fmaf(a, Wb[k * M + m], acc);
  }
  out[idx] = acc;
}

// ---- conv1 per-node transform as WMMA: for fixed r, x[:,r,:](64x200) @ W1full[r](200x32) ----
// grid (4 Mtiles, Rn), block 32. All dims exact multiples -> zero guards.
__global__ void xw1_wmma_kernel(const float* __restrict__ x, const float* __restrict__ Wf,
                                float* __restrict__ xw) {
  int tm   = blockIdx.x * 16;          // batch-row tile (64 rows total)
  int r    = blockIdx.y;
  int lane = threadIdx.x, half = lane >> 4, l = lane & 15;
  const float* Wr = Wf + (size_t)r * IND * D1;
  const float* Xm = x + (size_t)(tm + l) * Rn * IND + (size_t)r * IND;
  v8f acc0 = {}, acc1 = {};
#pragma unroll 2
  for (int k = 0; k < IND; k += 4) {
    int ka = k + half * 2;
    v2f a, b0, b1;
    a.x  = Xm[ka];                 a.y  = Xm[ka + 1];            // contiguous -> b64
    b0.x = Wr[ka * D1 + l];        b0.y = Wr[(ka + 1) * D1 + l];
    b1.x = Wr[ka * D1 + 16 + l];   b1.y = Wr[(ka + 1) * D1 + 16 + l];
    acc0 = wmma4(a, b0, acc0);
    acc1 = wmma4(a, b1, acc1);
  }
#pragma unroll
  for (int v = 0; v < 8; ++v) {
    int bRow = tm + v + half * 8;                 // batch index
    size_t base = ((size_t)bRow * Rn + r) * D1;
    xw[base + l]      = acc0[v];
    xw[base + 16 + l] = acc1[v];
  }
}

// ---- in-degree of adj rows: deg = max(nnz, 1) ----
__global__ void deg_adj_kernel(const float* __restrict__ adj, float* __restrict__ deg) {
  int idx = blockIdx.x * blockDim.x + threadIdx.x;
  if (idx >= Bz * Rn) return;
  const float* row = adj + (size_t)idx * Rn;
  int c = 0;
  for (int s = 0; s < Rn; ++s) c += (row[s] != 0.f);
  deg[idx] = (float)(c > 0 ? c : 1);
}

// ---- degree of (A+I)^2 rows with diagonal removed ----
__global__ void deg_msq_kernel(const float* __restrict__ msq, float* __restrict__ deg) {
  int idx = blockIdx.x * blockDim.x + threadIdx.x;
  if (idx >= Bz * Rn) return;
  int s = idx % Rn;
  const float* row = msq + (size_t)idx * Rn;
  int c = 0;
  for (int t = 0; t < Rn; ++t) c += (int)((t != s) && (row[t] != 0.f));
  deg[idx] = (float)(c > 0 ? c : 1);
}

// ---- WMMA batched GEMM: C[b](200x32) = A[b](200x200) @ Bm[b](200x32) ----
// 32-row strip per wave (2x2 tiles). Loads clamp row index (dupes of row 199
// are computed for rows 200..223 and discarded at store) -> branch-free loop.
__global__ void gemm_agg_kernel(const float* __restrict__ A, const float* __restrict__ Bm,
                                const float* __restrict__ deg, const float* __restrict__ bias,
                                float* __restrict__ C, int mode) {
  int b    = blockIdx.z;
  int tm   = blockIdx.x * 32;
  int lane = threadIdx.x, half = lane >> 4, l = lane & 15;
  const float* Ab = A  + (size_t)b * Rn * Rn;
  const float* Bb = Bm + (size_t)b * Rn * D1;
  int r0 = tm + l, r1 = r0 + 16;
  int r0c = r0 < Rn ? r0 : Rn - 1;
  int r1c = r1 < Rn ? r1 : Rn - 1;
  v8f acc00 = {}, acc01 = {}, acc10 = {}, acc11 = {};
#pragma unroll 2
  for (int k = 0; k < Rn; k += 4) {
    int ka = k + half * 2;
    v2f a0, a1, b0, b1;
    a0.x = Ab[r0c * Rn + ka];      a0.y = Ab[r0c * Rn + ka + 1];   // b64
    a1.x = Ab[r1c * Rn + ka];      a1.y = Ab[r1c * Rn + ka + 1];   // b64
    b0.x = Bb[ka * D1 + l];        b0.y = Bb[(ka + 1) * D1 + l];
    b1.x = Bb[ka * D1 + 16 + l];   b1.y = Bb[(ka + 1) * D1 + 16 + l];
    acc00 = wmma4(a0, b0, acc00);
    acc01 = wmma4(a0, b1, acc01);
    acc10 = wmma4(a1, b0, acc10);
    acc11 = wmma4(a1, b1, acc11);
  }
  float* Cb = C + (size_t)b * Rn * D1;
#pragma unroll
  for (int v = 0; v < 8; ++v) {
    int row0 = tm + v + half * 8;
    int row1 = row0 + 16;
    if (row0 < Rn) {
      float s = (mode == 1) ? deg[b * Rn + row0] : 1.f;
      float v0 = acc00[v], v1 = acc01[v];
      if (mode == 1) { v0 = v0 / s + bias[l]; v1 = v1 / s + bias[16 + l]; }
      Cb[row0 * D1 + l]      = v0;
      Cb[row0 * D1 + 16 + l] = v1;
    }
    if (row1 < Rn) {
      float s = (mode == 1) ? deg[b * Rn + row1] : 1.f;
      float v0 = acc10[v], v1 = acc11[v];
      if (mode == 1) { v0 = v0 / s + bias[l]; v1 = v1 / s + bias[16 + l]; }
      Cb[row1 * D1 + l]      = v0;
      Cb[row1 * D1 + 16 + l] = v1;
    }
  }
}

// ---- WMMA: Msq[b] = (adj[b]+I)^2, 200x200x200, 32x32 output block per wave ----
__global__ void msq_kernel(const float* __restrict__ adj, float* __restrict__ msq) {
  int b    = blockIdx.z;
  int tm   = blockIdx.x * 32, tn = blockIdx.y * 32;
  int lane = threadIdx.x, half = lane >> 4, l = lane & 15;
  const float* Ab = adj + (size_t)b * Rn * Rn;
  int r0 = tm + l, r1 = r0 + 16;
  int c0 = tn + l, c1 = c0 + 16;
  int r0c = r0 < Rn ? r0 : Rn - 1;
  int r1c = r1 < Rn ? r1 : Rn - 1;
  int c0c = c0 < Rn ? c0 : Rn - 1;
  int c1c = c1 < Rn ? c1 : Rn - 1;
  v8f acc00 = {}, acc01 = {}, acc10 = {}, acc11 = {};
#pragma unroll 2
  for (int k = 0; k < Rn; k += 4) {
    int ka = k + half * 2;
    v2f a0, a1, b0, b1;
    a0.x = Ab[r0c * Rn + ka]     + (r0c == ka     ? 1.f : 0.f);
    a0.y = Ab[r0c * Rn + ka + 1] + (r0c == ka + 1 ? 1.f : 0.f);
    a1.x = Ab[r1c * Rn + ka]     + (r1c == ka     ? 1.f : 0.f);
    a1.y = Ab[r1c * Rn + ka + 1] + (r1c == ka + 1 ? 1.f : 0.f);
    b0.x = Ab[ka * Rn + c0c]       + (ka == c0c     ? 1.f : 0.f);
    b0.y = Ab[(ka + 1) * Rn + c0c] + (ka + 1 == c0c ? 1.f : 0.f);
    b1.x = Ab[ka * Rn + c1c]       + (ka == c1c     ? 1.f : 0.f);
    b1.y = Ab[(ka + 1) * Rn + c1c] + (ka + 1 == c1c ? 1.f : 0.f);
    acc00 = wmma4(a0, b0, acc00);
    acc01 = wmma4(a0, b1, acc01);
    acc10 = wmma4(a1, b0, acc10);
    acc11 = wmma4(a1, b1, acc11);
  }
  float* Cb = msq + (size_t)b * Rn * Rn;
#pragma unroll
  for (int v = 0; v < 8; ++v) {
    int row0 = tm + v + half * 8;
    int row1 = row0 + 16;
    if (row0 < Rn) {
      if (c0 < Rn) Cb[row0 * Rn + c0] = acc00[v];
      if (c1 < Rn) Cb[row0 * Rn + c1] = acc01[v];
    }
    if (row1 < Rn) {
      if (c0 < Rn) Cb[row1 * Rn + c0] = acc10[v];
      if (c1 < Rn) Cb[row1 * Rn + c1] = acc11[v];
    }
  }
}

// ---- TopK score: sigmoid((feat . w) / ||w||) ----
__global__ void score_kernel(const float* __restrict__ feat, const float* __restrict__ w,
                             float* __restrict__ s_ws, float* __restrict__ s_out) {
  int idx = blockIdx.x * blockDim.x + threadIdx.x;
  if (idx >= Bz * Rn) return;
  float nrm = 0.f, d = 0.f;
#pragma unroll
  for (int o = 0; o < D1; ++o) { float t = w[o]; nrm = fmaf(t, t, nrm); }
  nrm = sqrtf(nrm);
  const float* f = feat + (size_t)idx * D1;
#pragma unroll
  for (int o = 0; o < D1; ++o) d = fmaf(f[o], w[o], d);
  float sc = 1.f / (1.f + expf(-d / nrm));
  s_ws[idx]  = sc;
  s_out[idx] = sc;
}

// ---- stable descending argsort by rank-counting (matches jnp.argsort(-score)) ----
__global__ void argsort_kernel(const float* __restrict__ score, int* __restrict__ perm_i,
                               float* __restrict__ perm_f) {
  __shared__ float s[Rn];
  int b = blockIdx.x;
  for (int r = threadIdx.x; r < Rn; r += blockDim.x) s[r] = score[b * Rn + r];
  __syncthreads();
  for (int r = threadIdx.x; r < Rn; r += blockDim.x) {
    float v = s[r];
    int rank = 0;
    for (int q = 0; q < Rn; ++q) {
      float u = s[q];
      rank += (int)((u > v) || ((u == v) && (q < r)));
    }
    perm_i[b * Rn + rank] = r;
    perm_f[b * Rn + rank] = (float)r;
  }
}

// ---- pooled, score-scaled features in permuted order ----
__global__ void pool_apply_kernel(const float* __restrict__ feat, const float* __restrict__ sc,
                                  const int* __restrict__ perm, float* __restrict__ hp) {
  int idx = blockIdx.x * blockDim.x + threadIdx.x;
  if (idx >= Bz * Rn * D1) return;
  int o = idx % D1;
  int j = (idx / D1) % Rn;
  int b = idx / (D1 * Rn);
  int p = perm[b * Rn + j];
  hp[idx] = feat[((size_t)b * Rn + p) * D1 + o] * sc[b * Rn + p];
}

// ---- conv2 per-node transform, scattered back to ORIGINAL node order ----
__global__ void xw2_kernel(const float* __restrict__ h1p, const float* __restrict__ W2f,
                           const int* __restrict__ perm, float* __restrict__ xw2o) {
  int br = blockIdx.x;           // b*Rn + i (pooled order)
  int b  = br / Rn;
  int o  = threadIdx.x;
  int r  = perm[br];             // original node id
  const float* h = h1p + (size_t)br * D1;
  const float* w = W2f + (size_t)r  * D1 * D2;
  float acc = 0.f;
#pragma unroll
  for (int d = 0; d < D1; ++d) acc = fmaf(h[d], w[d * D2 + o], acc);
  xw2o[((size_t)b * Rn + r) * D2 + o] = acc;
}

// ---- conv2 epilogue: gather + diagonal correction + /deg + bias ----
__global__ void conv2_epi_kernel(const float* __restrict__ rowdot, const float* __restrict__ xw2o,
                                 const float* __restrict__ msq, const float* __restrict__ deg2,
                                 const int* __restrict__ perm, const float* __restrict__ bias2,
                                 float* __restrict__ out2) {
  int idx = blockIdx.x * blockDim.x + threadIdx.x;
  if (idx >= Bz * Rn * D2) return;
  int o = idx % D2;
  int i = (idx / D2) % Rn;
  int b = idx / (D2 * Rn);
  int r = perm[b * Rn + i];
  size_t rr = (size_t)b * Rn + r;
  float diag = msq[((size_t)b * Rn + r) * Rn + r];
  out2[idx] = (rowdot[rr * D2 + o] - diag * xw2o[rr * D2 + o]) / deg2[b * Rn + r] + bias2[o];
}

// ---- readout: [max; mean] over nodes of feat*score (perm-invariant) ----
__global__ void readout_kernel(const float* __restrict__ feat, const float* __restrict__ sc,
                               float* __restrict__ z, int zoff) {
  int idx = blockIdx.x * blockDim.x + threadIdx.x;
  if (idx >= Bz * D1) return;
  int b = idx / D1, o = idx % D1;
  float mx = -__builtin_inff(), sm = 0.f;
  for (int n = 0; n < Rn; ++n) {
    float v = feat[((size_t)b * Rn + n) * D1 + o] * sc[b * Rn + n];
    mx = fmaxf(mx, v);
    sm += v;
  }
  z[b * 128 + zoff + o]      = mx;
  z[b * 128 + zoff + D1 + o] = sm * (1.f / Rn);
}

// ---- WMMA FC: C(64xN) = A(64xK) @ B(KxN) + bias; mode1: ReLU then g*x+bt ----
__global__ void fc_wmma_kernel(const float* __restrict__ A, const float* __restrict__ Bm,
                               const float* __restrict__ bias, const float* __restrict__ g,
                               const float* __restrict__ bt, float* __restrict__ C,
                               int K, int N, int mode) {
  int tm = blockIdx.x * 16, tn = blockIdx.y * 16;
  int lane = threadIdx.x, half = lane >> 4, l = lane & 15;
  int ra = tm + l, nb = tn + l;
  v8f c = {};
#pragma unroll 2
  for (int k = 0; k < K; k += 4) {
    int ka = k + half * 2;
    v2f av, bv;
    av.x = A[ra * K + ka];      av.y = A[ra * K + ka + 1];   // b64
    bv.x = Bm[ka * N + nb];     bv.y = Bm[(ka + 1) * N + nb];
    c = wmma4(av, bv, c);
  }
#pragma unroll
  for (int v = 0; v < 8; ++v) {
    int row = tm + v + half * 8, col = tn + l;
    float val = c[v] + bias[col];
    if (mode == 1) {
      val = val > 0.f ? val : 0.f;
      val = g[col] * val + bt[col];
    }
    C[row * N + col] = val;
  }
}

// ---- log_softmax over 256 + BN affine ----
__global__ void lsm_bn_kernel(const float* __restrict__ zin, const float* __restrict__ g,
                              const float* __restrict__ bt, float* __restrict__ zout) {
  __shared__ float red[D4];
  int b = blockIdx.x, t = threadIdx.x;
  float x = zin[b * D4 + t];
  red[t] = x;
  __syncthreads();
  for (int s = D4 / 2; s > 0; s >>= 1) {
    if (t < s) red[t] = fmaxf(red[t], red[t + s]);
    __syncthreads();
  }
  float mx = red[0];
  __syncthreads();
  red[t] = expf(x - mx);
  __syncthreads();
  for (int s = D4 / 2; s > 0; s >>= 1) {
    if (t < s) red[t] += red[t + s];
    __syncthreads();
  }
  float ls = logf(red[0]);
  zout[b * D4 + t] = g[t] * (x - mx - ls) + bt[t];
}

// ---- per-task heads: out[t,b,o] = z[b,:] . head_w[t,:,o] + head_b[t,o] ----
__global__ void heads_kernel(const float* __restrict__ z2, const float* __restrict__ hw,
                             const float* __restrict__ hb, float* __restrict__ out) {
  int idx = blockIdx.x * blockDim.x + threadIdx.x;
  if (idx >= Tt * Bz * D5) return;
  int o = idx % D5;
  int b = (idx / D5) % Bz;
  int t = idx / (D5 * Bz);
  float acc = hb[t * D5 + o];
  const float* zb = z2 + b * D4;
  const float* w  = hw + (size_t)t * D4 * D5 + o;
  for (int d = 0; d < D4; ++d) acc = fmaf(zb[d], w[d * D5], acc);
  out[idx] = acc;
}

extern "C" void kernel_launch(void* const* d_in, const int* in_sizes, int n_in,
                              void* d_out, int out_size, void* d_ws, size_t ws_size,
                              hipStream_t stream) {
  (void)in_sizes; (void)n_in; (void)out_size; (void)ws_size;
  const float* x     = (const float*)d_in[0];
  const float* adj   = (const float*)d_in[1];
  /* d_in[2] = pos: identity / derived from perm, not needed */
  const float* W1a   = (const float*)d_in[3];
  const float* W1b   = (const float*)d_in[4];
  const float* b1    = (const float*)d_in[5];
  const float* bias1 = (const float*)d_in[6];
  const float* pw1   = (const float*)d_in[7];
  const float* W2a   = (const float*)d_in[8];
  const float* W2b   = (const float*)d_in[9];
  const float* b2    = (const float*)d_in[10];
  const float* bias2 = (const float*)d_in[11];
  const float* pw2   = (const float*)d_in[12];
  const float* fc1w  = (const float*)d_in[13];
  const float* fc1b  = (const float*)d_in[14];
  const float* bn1g  = (const float*)d_in[15];
  const float* bn1b  = (const float*)d_in[16];
  const float* fc2w  = (const float*)d_in[17];
  const float* fc2b  = (const float*)d_in[18];
  const float* bn2g  = (const float*)d_in[19];
  const float* bn2b  = (const float*)d_in[20];
  const float* hw    = (const float*)d_in[21];
  const float* hb    = (const float*)d_in[22];

  float* out      = (float*)d_out;
  float* out_outs = out;                     // T*B*D5 = 1024
  float* out_s1   = out + 1024;              // 12800
  float* out_s2   = out + 1024 + 12800;      // 12800
  float* out_p1   = out + 1024 + 2 * 12800;  // 12800 (perm as float)
  float* out_p2   = out + 1024 + 3 * 12800;  // 12800

  float* ws = (float*)d_ws;
  size_t off = 0;
  float* W1full = ws + off; off += (size_t)Rn * IND * D1;   // 1,280,000
  float* W2full = ws + off; off += (size_t)Rn * D1 * D2;    //   204,800
  float* xw1    = ws + off; off += (size_t)Bz * Rn * D1;
  float* out1   = ws + off; off += (size_t)Bz * Rn * D1;
  float* s1ws   = ws + off; off += (size_t)Bz * Rn;
  int*   perm1i = (int*)(ws + off); off += (size_t)Bz * Rn;
  int*   perm2i = (int*)(ws + off); off += (size_t)Bz * Rn;
  float* deg1   = ws + off; off += (size_t)Bz * Rn;
  float* h1p    = ws + off; off += (size_t)Bz * Rn * D1;
  float* msq    = ws + off; off += (size_t)Bz * Rn * Rn;    // 2,560,000
  float* deg2   = ws + off; off += (size_t)Bz * Rn;
  float* xw2o   = ws + off; off += (size_t)Bz * Rn * D2;
  float* rowdot = ws + off; off += (size_t)Bz * Rn * D2;
  float* out2   = ws + off; off += (size_t)Bz * Rn * D2;
  float* s2ws   = ws + off; off += (size_t)Bz * Rn;
  float* zbuf   = ws + off; off += (size_t)Bz * 128;
  float* z1     = ws + off; off += (size_t)Bz * D3;
  float* z2raw  = ws + off; off += (size_t)Bz * D4;
  float* z2     = ws + off; off += (size_t)Bz * D4;

  const int NT = 256;
  hyper_kernel<<<(Rn * IND * D1 + NT - 1) / NT, NT, 0, stream>>>(W1a, W1b, b1, W1full, IND * D1);
  hyper_kernel<<<(Rn * D1 * D2 + NT - 1) / NT, NT, 0, stream>>>(W2a, W2b, b2, W2full, D1 * D2);
  xw1_wmma_kernel<<<dim3(4, Rn), 32, 0, stream>>>(x, W1full, xw1);
  deg_adj_kernel<<<(Bz * Rn + NT - 1) / NT, NT, 0, stream>>>(adj, deg1);
  gemm_agg_kernel<<<dim3(7, 1, Bz), 32, 0, stream>>>(adj, xw1, deg1, bias1, out1, 1);
  score_kernel<<<(Bz * Rn + NT - 1) / NT, NT, 0, stream>>>(out1, pw1, s1ws, out_s1);
  argsort_kernel<<<Bz, NT, 0, stream>>>(s1ws, perm1i, out_p1);
  pool_apply_kernel<<<(Bz * Rn * D1 + NT - 1) / NT, NT, 0, stream>>>(out1, s1ws, perm1i, h1p);
  msq_kernel<<<dim3(7, 7, Bz), 32, 0, stream>>>(adj, msq);
  deg_msq_kernel<<<(Bz * Rn + NT - 1) / NT, NT, 0, stream>>>(msq, deg2);
  xw2_kernel<<<Bz * Rn, D2, 0, stream>>>(h1p, W2full, perm1i, xw2o);
  gemm_agg_kernel<<<dim3(7, 1, Bz), 32, 0, stream>>>(msq, xw2o, deg2, bias2, rowdot, 0);
  conv2_epi_kernel<<<(Bz * Rn * D2 + NT - 1) / NT, NT, 0, stream>>>(rowdot, xw2o, msq, deg2,
                                                                    perm1i, bias2, out2);
  score_kernel<<<(Bz * Rn + NT - 1) / NT, NT, 0, stream>>>(out2, pw2, s2ws, out_s2);
  argsort_kernel<<<Bz, NT, 0, stream>>>(s2ws, perm2i, out_p2);
  readout_kernel<<<(Bz * D1 + NT - 1) / NT, NT, 0, stream>>>(out1, s1ws, zbuf, 0);
  readout_kernel<<<(Bz * D1 + NT - 1) / NT, NT, 0, stream>>>(out2, s2ws, zbuf, 64);
  fc_wmma_kernel<<<dim3(4, D3 / 16), 32, 0, stream>>>(zbuf, fc1w, fc1b, bn1g, bn1b, z1, 128, D3, 1);
  fc_wmma_kernel<<<dim3(4, D4 / 16), 32, 0, stream>>>(z1, fc2w, fc2b, nullptr, nullptr, z2raw, D3, D4, 0);
  lsm_bn_kernel<<<Bz, D4, 0, stream>>>(z2raw, bn2g, bn2b, z2);
  heads_kernel<<<(Tt * Bz * D5 + NT - 1) / NT, NT, 0, stream>>>(z2, hw, hb, out_outs);
}